// GBottleNeck_45217415692700
// MI455X (gfx1250) — compile-verified
//
#include <hip/hip_runtime.h>
#include <hip/hip_bf16.h>

// ---------------------------------------------------------------------------
// GCN bottleneck for MI455X (gfx1250, wave32).
// GEMMs use V_WMMA_F32_16X16X4_F32 (full fp32, matches reference numerics;
// GEMM is ~25 GFLOP total, far below the aggregation cost, so low-precision
// WMMA buys nothing here). Aggregation is edge-parallel with f32 global
// atomics (L2-resident working set: h = 25.6 MB << 192 MB L2).
// ---------------------------------------------------------------------------

typedef float v2f __attribute__((ext_vector_type(2)));
typedef float v8f __attribute__((ext_vector_type(8)));

#define N_NODES 50000
#define N_EDGES 800000
#define EHAT    (N_EDGES + N_NODES)
#define HID     128
#define IN_DIM  384
#define OUT_DIM 3
#define BLOCKS  6

__device__ __forceinline__ void atomic_add_f32(float* p, float v) {
    __hip_atomic_fetch_add(p, v, __ATOMIC_RELAXED, __HIP_MEMORY_SCOPE_AGENT);
}

// ---------------- degree / normalization ----------------------------------
__global__ void k_deg_init(float* deg) {                 // self-loop counts 1
    int i = blockIdx.x * blockDim.x + threadIdx.x;
    if (i < N_NODES) deg[i] = 1.0f;
}
__global__ void k_deg_accum(const int* __restrict__ ei, float* deg) {
    int e = blockIdx.x * blockDim.x + threadIdx.x;
    if (e < N_EDGES) atomic_add_f32(&deg[ei[N_EDGES + e]], 1.0f);
}
__global__ void k_rsqrt(float* dis) {                    // deg -> 1/sqrt(deg)
    int i = blockIdx.x * blockDim.x + threadIdx.x;
    if (i < N_NODES) dis[i] = rsqrtf(dis[i]);
}

// ---------------- fp32 WMMA GEMM: P[M,128] = A[M,K] @ W[K,128] --------------
// One block = 8 wave32s = 16(M) x 128(N) output tile. A-slab staged in LDS
// (pitch K+2 dwords: (K+2)%64==2 -> conflict-free b64 reads). Wave w owns
// columns [16w, 16w+16). WMMA fragment layout per ISA 7.12.2:
//   A 16x4 f32 : lanes 0-15 M=lane, v0=K+0,v1=K+1 ; lanes 16-31 v0=K+2,v1=K+3
//   B 4x16 f32 : lanes index N; low half lanes K={k,k+1}, high half K={k+2,k+3}
template <int K>
__global__ __launch_bounds__(256) void k_gemm_wmma(
        const float* __restrict__ A, const float* __restrict__ W,
        float* __restrict__ P) {
    __shared__ float As[16 * (K + 2)];
    const int tileM = blockIdx.x * 16;
    const int tid   = threadIdx.x;

    for (int i = tid; i < 16 * K; i += 256) {            // exact multiple: no divergence
        int r = i / K, c = i % K;
        As[r * (K + 2) + c] = A[(tileM + r) * K + c];
    }
    __syncthreads();

    const int wave = tid >> 5;
    const int lane = tid & 31;
    const int half = lane >> 4;                          // 0: K lo pair, 1: K hi pair
    const int lm   = lane & 15;
    const int col  = wave * 16 + lm;

    v8f c = {};
    for (int k = 0; k < K; k += 4) {
        const int kk = k + 2 * half;
        v2f a, b;
        a.x = As[lm * (K + 2) + kk];
        a.y = As[lm * (K + 2) + kk + 1];
        b.x = W[(kk + 0) * HID + col];
        b.y = W[(kk + 1) * HID + col];
        c = __builtin_amdgcn_wmma_f32_16x16x4_f32(false, a, false, b,
                                                  (short)0, c, false, false);
    }
    // C/D layout: VGPR i -> row i (lanes 0-15) / row 8+i (lanes 16-31)
    for (int i = 0; i < 8; ++i)
        P[(tileM + i + 8 * half) * HID + col] = c[i];
}

// ---------------- bias broadcast / relu / residual -------------------------
__global__ void k_fill_bias128(float* __restrict__ A, const float* __restrict__ b) {
    int i = blockIdx.x * blockDim.x + threadIdx.x;
    if (i < N_NODES * HID) A[i] = b[i & (HID - 1)];
}
__global__ void k_relu(const float* __restrict__ X, float* __restrict__ Y, int n) {
    int i = blockIdx.x * blockDim.x + threadIdx.x;
    if (i < n) Y[i] = fmaxf(X[i], 0.0f);
}
__global__ void k_resupdate(float* __restrict__ H, const float* __restrict__ R) {
    int i = blockIdx.x * blockDim.x + threadIdx.x;
    if (i < N_NODES * HID) H[i] = (H[i] + R[i]) * 0.5f;
}

// ---------------- edge aggregation, 128 channels: wave per edge ------------
__global__ __launch_bounds__(256) void k_agg128(
        const float* __restrict__ P, const int* __restrict__ ei,
        const float* __restrict__ dis, float* __restrict__ A) {
    int e = blockIdx.x * 8 + (threadIdx.x >> 5);
    if (e >= EHAT) return;
    const int lane = threadIdx.x & 31;
    int src, dst;
    if (e < N_EDGES) { src = ei[e]; dst = ei[N_EDGES + e]; }
    else             { src = dst = e - N_EDGES; }        // self-loop
    const float norm = dis[src] * dis[dst];
    const float4 v = ((const float4*)(P + (size_t)src * HID))[lane];
    float* out = A + (size_t)dst * HID + lane * 4;
    atomic_add_f32(out + 0, v.x * norm);
    atomic_add_f32(out + 1, v.y * norm);
    atomic_add_f32(out + 2, v.z * norm);
    atomic_add_f32(out + 3, v.w * norm);
}

// ---------------- final 128 -> 3 projection: wave per node -----------------
__global__ __launch_bounds__(256) void k_gemm3(
        const float* __restrict__ H, const float* __restrict__ Wf,
        float* __restrict__ P3) {
    int node = blockIdx.x * 8 + (threadIdx.x >> 5);
    if (node >= N_NODES) return;
    const int lane = threadIdx.x & 31;
    const float4 h = ((const float4*)(H + (size_t)node * HID))[lane];
    for (int o = 0; o < OUT_DIM; ++o) {
        float s = h.x * Wf[(lane * 4 + 0) * OUT_DIM + o]
                + h.y * Wf[(lane * 4 + 1) * OUT_DIM + o]
                + h.z * Wf[(lane * 4 + 2) * OUT_DIM + o]
                + h.w * Wf[(lane * 4 + 3) * OUT_DIM + o];
        for (int off = 16; off > 0; off >>= 1) s += __shfl_down(s, off, 32);
        if (lane == 0) P3[node * OUT_DIM + o] = s;
    }
}
__global__ void k_fill_bias3(float* __restrict__ A, const float* __restrict__ b) {
    int i = blockIdx.x * blockDim.x + threadIdx.x;
    if (i < N_NODES * OUT_DIM) A[i] = b[i % OUT_DIM];
}
__global__ void k_agg3(const float* __restrict__ P3, const int* __restrict__ ei,
                       const float* __restrict__ dis, float* __restrict__ A) {
    int e = blockIdx.x * blockDim.x + threadIdx.x;
    if (e >= EHAT) return;
    int src, dst;
    if (e < N_EDGES) { src = ei[e]; dst = ei[N_EDGES + e]; }
    else             { src = dst = e - N_EDGES; }
    const float norm = dis[src] * dis[dst];
    for (int c = 0; c < OUT_DIM; ++c)
        atomic_add_f32(&A[dst * OUT_DIM + c], P3[src * OUT_DIM + c] * norm);
}

// ---------------------------------------------------------------------------
extern "C" void kernel_launch(void* const* d_in, const int* in_sizes, int n_in,
                              void* d_out, int out_size, void* d_ws, size_t ws_size,
                              hipStream_t stream) {
    const float* x    = (const float*)d_in[0];
    const int*   ei   = (const int*)  d_in[1];
    const float* W0   = (const float*)d_in[2];
    const float* b0   = (const float*)d_in[3];
    const float* Wb1  = (const float*)d_in[4];
    const float* bb1  = (const float*)d_in[5];
    const float* Wb2  = (const float*)d_in[6];
    const float* bb2  = (const float*)d_in[7];
    const float* Wf   = (const float*)d_in[8];
    const float* bf   = (const float*)d_in[9];

    float* out = (float*)d_out;                    // [50000,3]
    float* H   = (float*)d_out + N_NODES * OUT_DIM; // [50000,128] (second output)

    float* ws   = (float*)d_ws;
    float* dis  = ws;                              // 50000 (deg -> rsqrt)
    float* buf0 = ws + 50048;                      // 50000*128
    float* buf1 = buf0 + N_NODES * HID;            // 50000*128

    const int T = 256;
    const int gNode = (N_NODES + T - 1) / T;
    const int gEdge = (N_EDGES + T - 1) / T;
    const int gNC   = (N_NODES * HID + T - 1) / T;
    const int gTile = N_NODES / 16;                // 3125, M % 16 == 0
    const int gAggW = (EHAT + 7) / 8;              // wave per edge
    const int gWarp = (N_NODES + 7) / 8;
    const int gE1   = (EHAT + T - 1) / T;
    const int gO3   = (N_NODES * OUT_DIM + T - 1) / T;

    // ---- normalization ----
    k_deg_init <<<gNode, T, 0, stream>>>(dis);
    k_deg_accum<<<gEdge, T, 0, stream>>>(ei, dis);
    k_rsqrt    <<<gNode, T, 0, stream>>>(dis);

    // ---- layer 0: h = relu(agg(x @ W0) + b0) ----
    k_gemm_wmma<IN_DIM><<<gTile, T, 0, stream>>>(x, W0, buf0);
    k_fill_bias128<<<gNC, T, 0, stream>>>(buf1, b0);
    k_agg128<<<gAggW, T, 0, stream>>>(buf0, ei, dis, buf1);
    k_relu<<<gNC, T, 0, stream>>>(buf1, H, N_NODES * HID);

    // ---- 6 residual blocks ----
    for (int i = 0; i < BLOCKS; ++i) {
        const float* W1 = Wb1 + (size_t)i * HID * HID;
        const float* c1 = bb1 + (size_t)i * HID;
        const float* W2 = Wb2 + (size_t)i * HID * HID;
        const float* c2 = bb2 + (size_t)i * HID;

        k_gemm_wmma<HID><<<gTile, T, 0, stream>>>(H, W1, buf0);
        k_fill_bias128<<<gNC, T, 0, stream>>>(buf1, c1);
        k_agg128<<<gAggW, T, 0, stream>>>(buf0, ei, dis, buf1);
        k_relu<<<gNC, T, 0, stream>>>(buf1, buf0, N_NODES * HID);

        k_gemm_wmma<HID><<<gTile, T, 0, stream>>>(buf0, W2, buf1);
        k_fill_bias128<<<gNC, T, 0, stream>>>(buf0, c2);
        k_agg128<<<gAggW, T, 0, stream>>>(buf1, ei, dis, buf0);
        k_relu<<<gNC, T, 0, stream>>>(buf0, buf0, N_NODES * HID);

        k_resupdate<<<gNC, T, 0, stream>>>(H, buf0);
    }

    // ---- output head: out = agg(H @ Wf) + bf ----
    k_gemm3<<<gWarp, T, 0, stream>>>(H, Wf, buf0);
    k_fill_bias3<<<gO3, T, 0, stream>>>(out, bf);
    k_agg3<<<gE1, T, 0, stream>>>(buf0, ei, dis, out);
}